// DenseGCN_27066883899810
// MI455X (gfx1250) — compile-verified
//
#include <hip/hip_runtime.h>
#include <hip/hip_bf16.h>

// ---------------------------------------------------------------------------
// DenseGCN forward for MI455X (gfx1250, wave32).
// GEMMs use V_WMMA_F32_16X16X4_F32 (fp32 WMMA) with BN+ReLU+norm folded into
// the A-operand load. Edge gather/scatter is fp32 atomics (L2-resident:
// 64MB features + 25.6MB indices << 192MB L2, HBM floor ~140us on edge
// traffic so the GEMMs must not add passes -> fully fused A operand).
// ---------------------------------------------------------------------------

#define NN 100000
#define NE 3200000
#define NG 256
#define NF0 32
#define GRW 32
#define NLAYERS 4
#define FINALF 160
#define BN_EPS 1e-5f

typedef float v2f __attribute__((ext_vector_type(2)));
typedef float v8f __attribute__((ext_vector_type(8)));

// ---------------- utility: zero fill ----------------
__global__ void k_zero(float* __restrict__ p, long long n) {
    long long i = (long long)blockIdx.x * blockDim.x + threadIdx.x;
    long long stride = (long long)gridDim.x * blockDim.x;
    for (; i < n; i += stride) p[i] = 0.0f;
}

// ---------------- edge weights + degrees ----------------
__global__ void k_edge_prep(const float* __restrict__ r,
                            const int* __restrict__ src,
                            const int* __restrict__ dst,
                            float* __restrict__ ew,
                            float* __restrict__ odeg,
                            float* __restrict__ ideg) {
    int e = blockIdx.x * blockDim.x + threadIdx.x;
    if (e >= NE) return;
    float x = r[3 * e + 0], y = r[3 * e + 1], z = r[3 * e + 2];
    ew[e] = expf(-(x * x + y * y + z * z) * (1.0f / 16.0f));  // LS=4 -> /16
    atomicAdd(&odeg[src[e]], 1.0f);
    atomicAdd(&ideg[dst[e]], 1.0f);
}

// deg -> rsqrt(clip(deg,1)) in place
__global__ void k_degree_norm(float* __restrict__ odeg, float* __restrict__ ideg) {
    int n = blockIdx.x * blockDim.x + threadIdx.x;
    if (n >= NN) return;
    odeg[n] = rsqrtf(fmaxf(odeg[n], 1.0f));
    ideg[n] = rsqrtf(fmaxf(ideg[n], 1.0f));
}

// ---------------- embedding: feat[:, 0:32] = a*W + b (AIF == 1) --------------
__global__ void k_embed(const float* __restrict__ af,
                        const float* __restrict__ Wemb,
                        const float* __restrict__ bemb,
                        float* __restrict__ feat) {
    int idx = blockIdx.x * blockDim.x + threadIdx.x;
    int node = idx >> 5, j = idx & 31;
    if (node >= NN) return;
    feat[(long long)node * FINALF + j] = af[node] * Wemb[j] + bemb[j];
}

// ---------------- in-place BN+ReLU on feat[:, 0:32] (bn0) --------------------
__global__ void k_bn0_apply(float* __restrict__ feat, const float* __restrict__ scale,
                            const float* __restrict__ shift) {
    int idx = blockIdx.x * blockDim.x + threadIdx.x;
    int node = idx >> 5, j = idx & 31;
    if (node >= NN) return;
    long long o = (long long)node * FINALF + j;
    feat[o] = fmaxf(feat[o] * scale[j] + shift[j], 0.0f);
}

// ---------------- per-column sum / sumsq (block-local then 1 atomic) ---------
// blockDim.x == ncols (multiple of 32); rows grid-strided over blockIdx.x
__global__ void k_col_stats(const float* __restrict__ x, int ncols,
                            float* __restrict__ sum, float* __restrict__ sumsq) {
    int c = threadIdx.x;
    float s = 0.0f, q = 0.0f;
    for (int rrow = blockIdx.x; rrow < NN; rrow += gridDim.x) {
        float v = x[(long long)rrow * FINALF + c];
        s += v;
        q += v * v;
    }
    atomicAdd(&sum[c], s);
    atomicAdd(&sumsq[c], q);
}

// scale = gamma*rsqrt(var+eps); shift = beta - mean*scale
__global__ void k_bn_coeffs(const float* __restrict__ sum, const float* __restrict__ sumsq,
                            const float* __restrict__ gamma, const float* __restrict__ beta,
                            float* __restrict__ scale, float* __restrict__ shift, int ncols) {
    int c = blockIdx.x * blockDim.x + threadIdx.x;
    if (c >= ncols) return;
    const float inv_n = 1.0f / (float)NN;
    float m = sum[c] * inv_n;
    float v = sumsq[c] * inv_n - m * m;  // biased variance, same as reference
    float sc = gamma[c] * rsqrtf(v + BN_EPS);
    scale[c] = sc;
    shift[c] = beta[c] - m * sc;
}

// ---------------- WMMA GEMM: hproj = relu(BN(feat[:, :fin])) .* nsrc @ W -----
// One wave computes a 16-row x 32-col tile via two f32 16x16x4 WMMA accums.
// ISA 7.12.2 layouts: A lane=M (l15), K split across lane halves + 2 VGPRs;
// B lane=N, same K striping; C/D: VGPR j -> M = j + 8*half, N = l15.
__global__ __launch_bounds__(128) void k_gemm_wmma(
    const float* __restrict__ feat, const float* __restrict__ scale,
    const float* __restrict__ shift, const float* __restrict__ nsrc,
    const float* __restrict__ W,  // [fin, 32] row-major
    float* __restrict__ hproj,    // [NN, 32]
    int fin) {
    int wave = (blockIdx.x * blockDim.x + threadIdx.x) >> 5;
    int lane = threadIdx.x & 31;
    int rowBase = wave * 16;
    if (rowBase >= NN) return;  // wave-uniform: EXEC stays all-1 for live waves
    int half = lane >> 4;       // K-half select
    int l15 = lane & 15;        // A: M index; B/C: N index
    int arow = rowBase + l15;
    float ns = nsrc[arow];
    const float* frow = feat + (long long)arow * FINALF;

    v8f acc0 = {};
    v8f acc1 = {};
    for (int k = 0; k < fin; k += 4) {
        int kk = k + 2 * half;
        // A operand: fused BN + ReLU + norm_src
        float x0 = fmaxf(frow[kk] * scale[kk] + shift[kk], 0.0f) * ns;
        float x1 = fmaxf(frow[kk + 1] * scale[kk + 1] + shift[kk + 1], 0.0f) * ns;
        v2f a;
        a.x = x0;
        a.y = x1;
        // B operand: lane = N column, K striped across the 2 VGPRs / lane halves
        v2f b0, b1;
        b0.x = W[kk * 32 + l15];
        b0.y = W[(kk + 1) * 32 + l15];
        b1.x = W[kk * 32 + 16 + l15];
        b1.y = W[(kk + 1) * 32 + 16 + l15];
#if __has_builtin(__builtin_amdgcn_wmma_f32_16x16x4_f32)
        acc0 = __builtin_amdgcn_wmma_f32_16x16x4_f32(false, a, false, b0, (short)0, acc0,
                                                     false, false);
        acc1 = __builtin_amdgcn_wmma_f32_16x16x4_f32(false, a, false, b1, (short)0, acc1,
                                                     false, false);
#else
        for (int j = 0; j < 8; ++j) {
            acc0[j] += a.x * b0.x + a.y * b0.y;
            acc1[j] += a.x * b1.x + a.y * b1.y;
        }
#endif
    }
    for (int j = 0; j < 8; ++j) {
        int rr = rowBase + j + 8 * half;
        hproj[(long long)rr * 32 + l15] = acc0[j];
        hproj[(long long)rr * 32 + 16 + l15] = acc1[j];
    }
}

// ---------------- edge gather * ew -> scatter-add ----------------------------
// 8 threads per edge, 4 contiguous floats each (float4 gather, 4 f32 atomics)
__global__ void k_edge_scatter(const int* __restrict__ src, const int* __restrict__ dst,
                               const float* __restrict__ ew,
                               const float* __restrict__ hproj,
                               float* __restrict__ agg) {
    int idx = blockIdx.x * blockDim.x + threadIdx.x;
    int e = idx >> 3;
    if (e >= NE) return;
    int jg = (idx & 7) * 4;
    float w = ew[e];
    const float4 v = *(const float4*)&hproj[(long long)src[e] * 32 + jg];
    float* a = &agg[(long long)dst[e] * 32 + jg];
    atomicAdd(a + 0, v.x * w);
    atomicAdd(a + 1, v.y * w);
    atomicAdd(a + 2, v.z * w);
    atomicAdd(a + 3, v.w * w);
}

// ---------------- layer finalize: feat[:, cb:cb+32] = agg*ndst + bias --------
__global__ void k_layer_finalize(const float* __restrict__ agg,
                                 const float* __restrict__ ndst,
                                 const float* __restrict__ bias,
                                 float* __restrict__ feat, int colbase) {
    int idx = blockIdx.x * blockDim.x + threadIdx.x;
    int node = idx >> 5, j = idx & 31;
    if (node >= NN) return;
    feat[(long long)node * FINALF + colbase + j] =
        agg[(long long)node * 32 + j] * ndst[node] + bias[j];
}

// ---------------- readout: per-graph sums of final BN+ReLU -------------------
__global__ void k_readout(const float* __restrict__ feat, const float* __restrict__ scale,
                          const float* __restrict__ shift, const int* __restrict__ gid,
                          float* __restrict__ gsum, float* __restrict__ gcnt) {
    int idx = blockIdx.x * blockDim.x + threadIdx.x;
    int node = idx / FINALF;
    int c = idx - node * FINALF;
    if (node >= NN) return;
    float v = fmaxf(feat[(long long)node * FINALF + c] * scale[c] + shift[c], 0.0f);
    int g = gid[node];
    atomicAdd(&gsum[g * FINALF + c], v);
    if (c == 0) atomicAdd(&gcnt[g], 1.0f);
}

// ---------------- final FC: out[g] = mean @ W_fc + b_fc ----------------------
__global__ void k_final_fc(const float* __restrict__ gsum, const float* __restrict__ gcnt,
                           const float* __restrict__ Wfc, const float* __restrict__ bfc,
                           float* __restrict__ out) {
    int g = blockIdx.x * blockDim.x + threadIdx.x;
    if (g >= NG) return;
    float inv = 1.0f / fmaxf(gcnt[g], 1.0f);
    float acc = 0.0f;
    for (int c = 0; c < FINALF; ++c) acc += gsum[g * FINALF + c] * Wfc[c];
    out[g] = acc * inv + bfc[0];
}

// ---------------------------------------------------------------------------
extern "C" void kernel_launch(void* const* d_in, const int* in_sizes, int n_in,
                              void* d_out, int out_size, void* d_ws, size_t ws_size,
                              hipStream_t stream) {
    // --- inputs (setup_inputs dict order; lists expand to per-element entries)
    const float* af = (const float*)d_in[0];    // atom_features [NN,1]
    const float* r = (const float*)d_in[1];     // r [NE,3]
    const int* src = (const int*)d_in[2];       // [NE]
    const int* dst = (const int*)d_in[3];       // [NE]
    const int* gid = (const int*)d_in[4];       // [NN]
    const float* Wemb = (const float*)d_in[5];  // [1,32]
    const float* bemb = (const float*)d_in[6];  // [32]
    const float* convW[NLAYERS];
    const float* convB[NLAYERS];
    const float* bnG[NLAYERS];
    const float* bnB[NLAYERS];
    const float *bn0g, *bn0b, *bnfg, *bnfb, *Wfc, *bfc;
    if (n_in >= 29) {  // lists expanded into separate inputs
        bn0g = (const float*)d_in[7];
        bn0b = (const float*)d_in[8];
        for (int i = 0; i < NLAYERS; ++i) {
            convW[i] = (const float*)d_in[9 + i];
            convB[i] = (const float*)d_in[13 + i];
            bnG[i] = (const float*)d_in[17 + i];
            bnB[i] = (const float*)d_in[21 + i];
        }
        bnfg = (const float*)d_in[25];
        bnfb = (const float*)d_in[26];
        Wfc = (const float*)d_in[27];
        bfc = (const float*)d_in[28];
    } else {  // lists flattened/concatenated into one array each
        bn0g = (const float*)d_in[7];
        bn0b = (const float*)d_in[8];
        const float* wcat = (const float*)d_in[9];
        const float* bcat = (const float*)d_in[10];
        const float* gcat = (const float*)d_in[11];
        const float* betcat = (const float*)d_in[12];
        size_t woff = 0, voff = 0;
        for (int i = 0; i < NLAYERS; ++i) {
            int fin = NF0 + i * GRW;
            convW[i] = wcat + woff;
            woff += (size_t)fin * GRW;
            convB[i] = bcat + (size_t)i * GRW;
            bnG[i] = gcat + voff;
            bnB[i] = betcat + voff;
            voff += fin;
        }
        bnfg = (const float*)d_in[13];
        bnfb = (const float*)d_in[14];
        Wfc = (const float*)d_in[15];
        bfc = (const float*)d_in[16];
    }
    float* out = (float*)d_out;

    // --- workspace carve-up (all fp32), 256B aligned; ~103 MB total
    char* ws = (char*)d_ws;
    size_t off = 0;
    auto carve = [&](size_t nfloats) {
        float* p = (float*)(ws + off);
        off = (off + nfloats * sizeof(float) + 255) & ~(size_t)255;
        return p;
    };
    float* feat = carve((size_t)NN * FINALF);  // 64 MB  concatenated features
    float* hproj = carve((size_t)NN * 32);     // 12.8 MB
    float* agg = carve((size_t)NN * 32);       // 12.8 MB
    float* ew = carve((size_t)NE);             // 12.8 MB
    float* nsrc = carve(NN);                   // out-degree -> rsqrt
    float* ndst = carve(NN);                   // in-degree  -> rsqrt
    float* colsum = carve(FINALF);
    float* colsq = carve(FINALF);
    float* scale = carve(FINALF);
    float* shift = carve(FINALF);
    float* gsum = carve((size_t)NG * FINALF);
    float* gcnt = carve(NG);
    (void)ws_size;

    const int ZB = 2048, TB = 256;
    const int nodeJ_blocks = (int)(((long long)NN * 32 + TB - 1) / TB);

    // 1) degrees + edge weights
    k_zero<<<ZB, TB, 0, stream>>>(nsrc, NN);
    k_zero<<<ZB, TB, 0, stream>>>(ndst, NN);
    k_edge_prep<<<(NE + TB - 1) / TB, TB, 0, stream>>>(r, src, dst, ew, nsrc, ndst);
    k_degree_norm<<<(NN + TB - 1) / TB, TB, 0, stream>>>(nsrc, ndst);

    // 2) embedding -> feat[:, 0:32], then bn0 + ReLU applied IN PLACE so
    //    feat[:,0:32] == features[0] of the reference (the activated h).
    k_embed<<<nodeJ_blocks, TB, 0, stream>>>(af, Wemb, bemb, feat);
    k_zero<<<1, TB, 0, stream>>>(colsum, FINALF);
    k_zero<<<1, TB, 0, stream>>>(colsq, FINALF);
    k_col_stats<<<256, NF0, 0, stream>>>(feat, NF0, colsum, colsq);
    k_bn_coeffs<<<1, TB, 0, stream>>>(colsum, colsq, bn0g, bn0b, scale, shift, NF0);
    k_bn0_apply<<<nodeJ_blocks, TB, 0, stream>>>(feat, scale, shift);

    // 3) dense layers: layer i applies bn_gammas[i]/bn_betas[i] to the concat
    //    (fused into the WMMA A-operand), projects with convW[i], then
    //    gather/scale/scatter over edges, finalize with norm_dst + bias.
    const int gemm_blocks = ((NN / 16) * 32 + 127) / 128;  // 1 wave per 16-row tile
    for (int i = 0; i < NLAYERS; ++i) {
        int fin = NF0 + i * GRW;

        // BN stats over feat[:, :fin]
        k_zero<<<1, TB, 0, stream>>>(colsum, FINALF);
        k_zero<<<1, TB, 0, stream>>>(colsq, FINALF);
        k_col_stats<<<256, fin, 0, stream>>>(feat, fin, colsum, colsq);
        k_bn_coeffs<<<1, TB, 0, stream>>>(colsum, colsq, bnG[i], bnB[i], scale, shift, fin);

        // fused BN+ReLU+norm GEMM (f32 WMMA)
        k_gemm_wmma<<<gemm_blocks, 128, 0, stream>>>(feat, scale, shift, nsrc, convW[i],
                                                     hproj, fin);

        // scatter
        k_zero<<<ZB, TB, 0, stream>>>(agg, (long long)NN * 32);
        k_edge_scatter<<<(int)(((long long)NE * 8 + TB - 1) / TB), TB, 0, stream>>>(
            src, dst, ew, hproj, agg);
        k_layer_finalize<<<nodeJ_blocks, TB, 0, stream>>>(agg, ndst, convB[i], feat, fin);
    }

    // 4) final BN over all 160 cols
    k_zero<<<1, TB, 0, stream>>>(colsum, FINALF);
    k_zero<<<1, TB, 0, stream>>>(colsq, FINALF);
    k_col_stats<<<256, FINALF, 0, stream>>>(feat, FINALF, colsum, colsq);
    k_bn_coeffs<<<1, TB, 0, stream>>>(colsum, colsq, bnfg, bnfb, scale, shift, FINALF);

    // 5) readout + FC
    k_zero<<<1, TB, 0, stream>>>(gcnt, NG);
    k_zero<<<ZB, TB, 0, stream>>>(gsum, (long long)NG * FINALF);
    k_readout<<<(int)(((long long)NN * FINALF + TB - 1) / TB), TB, 0, stream>>>(
        feat, scale, shift, gid, gsum, gcnt);
    k_final_fc<<<1, TB, 0, stream>>>(gsum, gcnt, Wfc, bfc, out);
    (void)out_size;
    (void)in_sizes;
}